// ContiFormerBackbone_3212635537827
// MI455X (gfx1250) — compile-verified
//
#include <hip/hip_runtime.h>
#include <hip/hip_bf16.h>

// ---------------------------------------------------------------------------
// ContiFormer backbone for MI455X (gfx1250, wave32).
// Matmuls: V_WMMA_F32_16X16X4_F32 (full f32 matrix pipe), dual accumulator
// chains to hide WMMA->WMMA RAW hazard NOPs.
// Operand staging: TENSOR_LOAD_TO_LDS (TDM DMA) with hardware LDS padding.
// ---------------------------------------------------------------------------

typedef float v2f __attribute__((ext_vector_type(2)));
typedef float v8f __attribute__((ext_vector_type(8)));
typedef unsigned int u32x4 __attribute__((ext_vector_type(4)));
typedef int i32x4 __attribute__((ext_vector_type(4)));
typedef int i32x8 __attribute__((ext_vector_type(8)));

static constexpr int B_  = 16;
static constexpr int T_  = 1024;
static constexpr int D_  = 512;
static constexpr int H_  = 128;
static constexpr int NH_ = 4;
static constexpr int DH_ = 32;
static constexpr int TOK = B_ * T_;          // 16384 rows

__device__ __forceinline__ v8f vzero8() {
  v8f z = {0.f,0.f,0.f,0.f,0.f,0.f,0.f,0.f};
  return z;
}

// D(16x16,f32) += A(16x4,f32) * B(4x16,f32)
__device__ __forceinline__ v8f wmma4(v2f a, v2f b, v8f c) {
  return __builtin_amdgcn_wmma_f32_16x16x4_f32(
      /*neg_a=*/false, a, /*neg_b=*/false, b,
      /*c_mod=*/(short)0, c, /*reuse_a=*/false, /*reuse_b=*/false);
}

__device__ __forceinline__ float wave_sum(float v) {
  for (int off = 16; off > 0; off >>= 1) v += __shfl_xor(v, off, 32);
  return v;
}
__device__ __forceinline__ float wave_max(float v) {
  for (int off = 16; off > 0; off >>= 1) v = fmaxf(v, __shfl_xor(v, off, 32));
  return v;
}

// ---------------------------------------------------------------------------
// TDM: DMA a 2-D f32 tile (tile_d1 rows x tile_d0 elems, global row stride
// stride_d0 elems) into LDS at lds_byte_off, inserting pad_amount(code)
// DWORDs of LDS padding every pad_interval(code) DWORDs (per-row padding
// when the interval equals the row length).
// ---------------------------------------------------------------------------
__device__ __forceinline__ void tdm_load_2d(const void* gsrc,
                                            unsigned lds_byte_off,
                                            unsigned tile_d0, unsigned tile_d1,
                                            unsigned stride_d0,
                                            unsigned pad_interval_code,
                                            unsigned pad_amount_code) {
  unsigned long long ga = (unsigned long long)(uintptr_t)gsrc;
  u32x4 g0;
  g0.x = 1u;                                        // count=1, user descriptor
  g0.y = lds_byte_off;                              // lds_addr
  g0.z = (unsigned)ga;                              // global_addr[31:0]
  g0.w = ((unsigned)(ga >> 32) & 0x1FFFFFFu) | (2u << 30);  // addr[56:32]|type=2
  i32x8 g1;
  g1[0] = (int)((2u << 16)                          // data_size = 4B
              | (1u << 20)                          // pad_enable
              | (pad_interval_code << 22)
              | (pad_amount_code << 25));
  g1[1] = (int)((stride_d0 & 0xFFFFu) << 16);       // tensor_dim0[15:0]
  g1[2] = (int)(((stride_d0 >> 16) & 0xFFFFu)       // tensor_dim0[31:16]
              | ((tile_d1 & 0xFFFFu) << 16));       // tensor_dim1[15:0]
  g1[3] = (int)((tile_d0 & 0xFFFFu) << 16);         // tensor_dim1 hi=0 | tile_dim0
  g1[4] = (int)(tile_d1 & 0xFFFFu);                 // tile_dim1 | tile_dim2=0
  g1[5] = (int)stride_d0;                           // tensor_dim0_stride[31:0]
  g1[6] = 0;                                        // stride hi | dim1_stride lo
  g1[7] = 0;
  i32x4 z4 = {0,0,0,0};
#if __clang_major__ >= 23
  i32x8 z8 = {0,0,0,0,0,0,0,0};
  __builtin_amdgcn_tensor_load_to_lds(g0, g1, z4, z4, z8, 0);
#else
  __builtin_amdgcn_tensor_load_to_lds(g0, g1, z4, z4, 0);
#endif
}

// ---------------------------------------------------------------------------
// 1) dm[bt] = mean_D(time_delta[bt, :])     one wave per row
// ---------------------------------------------------------------------------
__global__ void dm_kernel(const float* __restrict__ td, float* __restrict__ dm) {
  int row  = (blockIdx.x * blockDim.x + threadIdx.x) >> 5;
  int lane = threadIdx.x & 31;
  const float* p = td + (size_t)row * D_;
  float s = 0.f;
  for (int i = lane; i < D_; i += 32) s += p[i];
  s = wave_sum(s);
  if (lane == 0) dm[row] = s * (1.0f / (float)D_);
}

// ---------------------------------------------------------------------------
// 2) base = x @ W_in + b_in + te ;  decay = exp(-relu(dm*decay_w + decay_b))
//    WMMA f32 GEMM, M-tile 16, N = 128, K = 512 staged by TDM in 128 chunks.
// ---------------------------------------------------------------------------
__global__ void input_gemm(const float* __restrict__ x,
                           const float* __restrict__ Win,
                           const float* __restrict__ b_in,
                           const float* __restrict__ vt,
                           const float* __restrict__ dm,
                           const float* __restrict__ te_freq,
                           const float* __restrict__ te_phase,
                           const float* __restrict__ te_dw,
                           const float* __restrict__ te_db,
                           const float* __restrict__ decay_w,
                           const float* __restrict__ decay_b,
                           float* __restrict__ base,
                           float* __restrict__ decay) {
  __shared__ float As[16][132];            // 128-wide K chunk, stride 132 via TDM pad
  const int wave = threadIdx.x >> 5;
  const int lane = threadIdx.x & 31;
  const int m0   = blockIdx.x << 4;
  const int col  = (wave << 4) + (lane & 15);
  const int r    = lane & 15;
  const int kb0  = (lane >> 4) << 1;

  v8f acc0 = vzero8(), acc1 = vzero8();
  for (int k0 = 0; k0 < D_; k0 += 128) {
    __syncthreads();
    if (threadIdx.x < 32) {   // wave 0 drives the TDM
      tdm_load_2d(x + (size_t)m0 * D_ + k0, 0u,
                  /*tile_d0=*/128, /*tile_d1=*/16, /*stride=*/D_,
                  /*pad: +4 DW every 128 DW*/ 6, 3);
      __builtin_amdgcn_s_wait_tensorcnt(0);
    }
    __syncthreads();
    for (int kk = 0; kk < 128; kk += 8) {
      int kg0 = k0 + kk + kb0;
      v2f a0; a0.x = As[r][kk + kb0];     a0.y = As[r][kk + kb0 + 1];
      v2f b0; b0.x = Win[(size_t)kg0 * H_ + col];
              b0.y = Win[(size_t)(kg0 + 1) * H_ + col];
      acc0 = wmma4(a0, b0, acc0);
      int kg1 = kg0 + 4;
      v2f a1; a1.x = As[r][kk + 4 + kb0]; a1.y = As[r][kk + 4 + kb0 + 1];
      v2f b1; b1.x = Win[(size_t)kg1 * H_ + col];
              b1.y = Win[(size_t)(kg1 + 1) * H_ + col];
      acc1 = wmma4(a1, b1, acc1);
    }
  }
  v8f acc = acc0 + acc1;
  const float bi = b_in[col];
  const float f  = te_freq[col], ph = te_phase[col];
  const float dw = te_dw[col],  db = te_db[col];
  const float cw = decay_w[col], cb = decay_b[col];
  for (int j = 0; j < 8; ++j) {
    int m = m0 + j + ((lane >> 4) << 3);
    float vtm = vt[m], dmm = dm[m];
    float te  = __sinf(vtm * f + ph) + dmm * dw + db;
    base[(size_t)m * H_ + col] = acc[j] + bi + te;
    float dz = dmm * cw + cb;
    decay[(size_t)m * H_ + col] = __expf(-fmaxf(dz, 0.f));
  }
}

// ---------------------------------------------------------------------------
// 3) sequential decay scan: one thread per (b, h) channel
// ---------------------------------------------------------------------------
__global__ void scan_kernel(const float* __restrict__ base,
                            const float* __restrict__ decay,
                            const int* __restrict__ lengths,
                            float* __restrict__ h) {
  int idx = blockIdx.x * blockDim.x + threadIdx.x;   // [0, B*H)
  int b   = idx / H_;
  int hh  = idx - b * H_;
  int len = lengths[b];
  const float* bp = base  + (size_t)b * T_ * H_ + hh;
  const float* dp = decay + (size_t)b * T_ * H_ + hh;
  float* hp       = h     + (size_t)b * T_ * H_ + hh;
  float s = 0.f;
  for (int t = 0; t < T_; ++t) {
    if (t < len) { s = dp[(size_t)t * H_] * s + bp[(size_t)t * H_];
                   hp[(size_t)t * H_] = s; }
    else         { hp[(size_t)t * H_] = 0.f; }
  }
}

// ---------------------------------------------------------------------------
// 4) LayerNorm over H=128: one wave per row, 4 elements / lane
// ---------------------------------------------------------------------------
__global__ void ln_kernel(const float* __restrict__ h,
                          const float* __restrict__ g,
                          const float* __restrict__ bta,
                          float* __restrict__ hn) {
  int row  = (blockIdx.x * blockDim.x + threadIdx.x) >> 5;
  int lane = threadIdx.x & 31;
  const float* p = h + ((size_t)row << 7);
  float v0 = p[lane], v1 = p[lane + 32], v2 = p[lane + 64], v3 = p[lane + 96];
  float mu = wave_sum(v0 + v1 + v2 + v3) * (1.f / 128.f);
  float d0 = v0 - mu, d1 = v1 - mu, d2 = v2 - mu, d3 = v3 - mu;
  float var = wave_sum(d0*d0 + d1*d1 + d2*d2 + d3*d3) * (1.f / 128.f);
  float inv = rsqrtf(var + 1e-5f);
  float* q = hn + ((size_t)row << 7);
  q[lane]      = d0 * inv * g[lane]      + bta[lane];
  q[lane + 32] = d1 * inv * g[lane + 32] + bta[lane + 32];
  q[lane + 64] = d2 * inv * g[lane + 64] + bta[lane + 64];
  q[lane + 96] = d3 * inv * g[lane + 96] + bta[lane + 96];
}

// ---------------------------------------------------------------------------
// 5) Generic WMMA GEMM: C[M,128] = A[M,128] @ W[128,128] + bias (+ residual)
//    A tile (16x128, contiguous) staged by TDM with +4DW/128DW padding.
// ---------------------------------------------------------------------------
__global__ void gemm128(const float* __restrict__ A,
                        const float* __restrict__ W,
                        const float* __restrict__ bias,
                        const float* __restrict__ residual,
                        float* __restrict__ C) {
  __shared__ float As[16][132];
  const int wave = threadIdx.x >> 5;
  const int lane = threadIdx.x & 31;
  const int m0   = blockIdx.x << 4;
  const int col  = (wave << 4) + (lane & 15);
  const int r    = lane & 15;
  const int kb0  = (lane >> 4) << 1;

  if (threadIdx.x < 32) {
    tdm_load_2d(A + ((size_t)m0 << 7), 0u,
                /*tile_d0=*/128, /*tile_d1=*/16, /*stride=*/128,
                /*pad: +4 DW every 128 DW*/ 6, 3);
    __builtin_amdgcn_s_wait_tensorcnt(0);
  }
  __syncthreads();

  v8f acc0 = vzero8(), acc1 = vzero8();
  for (int kk = 0; kk < 128; kk += 8) {
    v2f a0; a0.x = As[r][kk + kb0];     a0.y = As[r][kk + kb0 + 1];
    v2f b0; b0.x = W[((size_t)(kk + kb0) << 7) + col];
            b0.y = W[((size_t)(kk + kb0 + 1) << 7) + col];
    acc0 = wmma4(a0, b0, acc0);
    v2f a1; a1.x = As[r][kk + 4 + kb0]; a1.y = As[r][kk + 4 + kb0 + 1];
    v2f b1; b1.x = W[((size_t)(kk + 4 + kb0) << 7) + col];
            b1.y = W[((size_t)(kk + 5 + kb0) << 7) + col];
    acc1 = wmma4(a1, b1, acc1);
  }
  v8f acc = acc0 + acc1;
  const float bi = bias[col];
  for (int j = 0; j < 8; ++j) {
    int m = m0 + j + ((lane >> 4) << 3);
    float v = acc[j] + bi;
    if (residual) v += residual[((size_t)m << 7) + col];
    C[((size_t)m << 7) + col] = v;
  }
}

// ---------------------------------------------------------------------------
// 6) Attention: one workgroup per (b, head, 16-row q tile).
//    QK^T via WMMA into 16x1024 LDS scores, +dt*w_rel bias, key mask,
//    row softmax (wave reductions), P@V via WMMA (K split across waves,
//    merged with ds_add_f32 atomics). Q tile staged by TDM.
// ---------------------------------------------------------------------------
static constexpr int SROW = 1028;            // padded score row stride
static constexpr int QSTR = 36;              // padded Q tile row stride
static constexpr int ATTN_LDS_FLOATS = 16*SROW + 16*QSTR + 16*32 + 16 + 16;

__global__ void attn_kernel(const float* __restrict__ Q,
                            const float* __restrict__ K,
                            const float* __restrict__ V,
                            const float* __restrict__ vt,
                            const int* __restrict__ lengths,
                            const float* __restrict__ w_rel,
                            float* __restrict__ O) {
  extern __shared__ float sm[];
  float* S    = sm;                     // [16][SROW]
  float* Qt   = S   + 16 * SROW;        // [16][QSTR]
  float* Ot   = Qt  + 16 * QSTR;        // [16][32] accumulators
  float* rinv = Ot  + 16 * 32;          // [16] 1/rowsum
  float* vtq  = rinv + 16;              // [16] q visit times

  const int qt   = blockIdx.x & 63;
  const int head = (blockIdx.x >> 6) & 3;
  const int b    = blockIdx.x >> 8;
  const int wave = threadIdx.x >> 5;
  const int lane = threadIdx.x & 31;
  const int len  = lengths[b];
  const float wrel = w_rel[head];
  const size_t baseP = (size_t)b * T_ * H_ + head * DH_;

  // stage Q tile via TDM (rows of 32 DW padded to 36), zero accumulators
  if (threadIdx.x < 32) {
    tdm_load_2d(Q + baseP + (size_t)(qt * 16) * H_,
                (unsigned)(16u * SROW * sizeof(float)),
                /*tile_d0=*/DH_, /*tile_d1=*/16, /*stride=*/H_,
                /*pad: +4 DW every 32 DW*/ 4, 3);
    __builtin_amdgcn_s_wait_tensorcnt(0);
  }
  for (int i = threadIdx.x; i < 16 * DH_; i += 256) Ot[i] = 0.f;
  if (threadIdx.x < 16) vtq[threadIdx.x] = vt[b * T_ + qt * 16 + threadIdx.x];
  __syncthreads();

  const int r   = lane & 15;
  const int kb0 = (lane >> 4) << 1;

  // ---- scores: S = Q @ K^T / sqrt(DH) + dt*w_rel, masked -----------------
  for (int kt = wave; kt < 64; kt += 8) {
    v8f acc0 = vzero8(), acc1 = vzero8();
    int key = kt * 16 + r;
    const float* kp = K + baseP + (size_t)key * H_;
    for (int kk = 0; kk < DH_; kk += 8) {
      v2f a0; a0.x = Qt[r * QSTR + kk + kb0];     a0.y = Qt[r * QSTR + kk + kb0 + 1];
      v2f b0; b0.x = kp[kk + kb0];                b0.y = kp[kk + kb0 + 1];
      acc0 = wmma4(a0, b0, acc0);
      v2f a1; a1.x = Qt[r * QSTR + kk + 4 + kb0]; a1.y = Qt[r * QSTR + kk + 4 + kb0 + 1];
      v2f b1; b1.x = kp[kk + 4 + kb0];            b1.y = kp[kk + 4 + kb0 + 1];
      acc1 = wmma4(a1, b1, acc1);
    }
    v8f acc = acc0 + acc1;
    float vtk = vt[b * T_ + key];
    bool kvalid = key < len;
    for (int j = 0; j < 8; ++j) {
      int m = j + ((lane >> 4) << 3);
      float s = acc[j] * 0.17677669529663687f + (vtq[m] - vtk) * wrel;
      S[m * SROW + kt * 16 + r] = kvalid ? s : -1e9f;
    }
  }
  __syncthreads();

  // ---- softmax per row (2 rows per wave) ---------------------------------
  for (int i = 0; i < 2; ++i) {
    int row = wave * 2 + i;
    float* sr = S + row * SROW;
    float mx = -3.0e38f;
    for (int c = lane; c < T_; c += 32) mx = fmaxf(mx, sr[c]);
    mx = wave_max(mx);
    float sum = 0.f;
    for (int c = lane; c < T_; c += 32) {
      float e = __expf(sr[c] - mx);
      sr[c] = e; sum += e;
    }
    sum = wave_sum(sum);
    if (lane == 0) rinv[row] = 1.0f / sum;
  }
  __syncthreads();

  // ---- O = P @ V : 2 n-tiles x 4 K-chunks over the 8 waves ---------------
  {
    const int ntile  = wave & 1;
    const int kchunk = wave >> 1;
    const int n = ntile * 16 + r;
    v8f acc0 = vzero8(), acc1 = vzero8();
    const float* vp = V + baseP + n;
    for (int kk = 0; kk < 256; kk += 8) {
      int kb = kchunk * 256 + kk + kb0;
      v2f a0; a0.x = S[r * SROW + kb];     a0.y = S[r * SROW + kb + 1];
      v2f b0; b0.x = vp[(size_t)kb * H_];  b0.y = vp[(size_t)(kb + 1) * H_];
      acc0 = wmma4(a0, b0, acc0);
      int kb1 = kb + 4;
      v2f a1; a1.x = S[r * SROW + kb1];    a1.y = S[r * SROW + kb1 + 1];
      v2f b1; b1.x = vp[(size_t)kb1 * H_]; b1.y = vp[(size_t)(kb1 + 1) * H_];
      acc1 = wmma4(a1, b1, acc1);
    }
    v8f acc = acc0 + acc1;
    for (int j = 0; j < 8; ++j) {
      int m = j + ((lane >> 4) << 3);
      atomicAdd(&Ot[m * 32 + n], acc[j]);
    }
  }
  __syncthreads();

  // ---- normalize + write out --------------------------------------------
  for (int i = threadIdx.x; i < 16 * DH_; i += 256) {
    int rr = i >> 5, dd = i & 31;
    O[baseP + (size_t)(qt * 16 + rr) * H_ + dd] = Ot[i] * rinv[rr];
  }
}

// ---------------------------------------------------------------------------
// Host-side orchestration
// ---------------------------------------------------------------------------
extern "C" void kernel_launch(void* const* d_in, const int* in_sizes, int n_in,
                              void* d_out, int out_size, void* d_ws, size_t ws_size,
                              hipStream_t stream) {
  (void)in_sizes; (void)n_in; (void)out_size; (void)ws_size;

  const float* x   = (const float*)d_in[0];     // [B,T,D]
  const float* vt  = (const float*)d_in[1];     // [B,T]
  const float* td  = (const float*)d_in[2];     // [B,T,D]
  const int* lens  = (const int*)  d_in[3];     // [B]
  // d_in[4] = missing_mask (unused)
  const float* Win = (const float*)d_in[5];
  const float* bin = (const float*)d_in[6];
  const float* tef = (const float*)d_in[7];
  const float* tep = (const float*)d_in[8];
  const float* tdw = (const float*)d_in[9];
  const float* tdb = (const float*)d_in[10];
  const float* dcw = (const float*)d_in[11];
  const float* dcb = (const float*)d_in[12];
  // per layer (insertion order): Wq Wk Wv Wo bq bk bv bo w_rel ln_g ln_b
  const int LBASE = 13, LSTRIDE = 11;

  float* ws    = (float*)d_ws;
  const size_t R = (size_t)TOK * H_;            // 2M floats per region
  float* dm    = ws;
  float* base  = ws + 32768;
  float* decay = base  + R;
  float* h     = decay + R;
  float* qb    = h     + R;
  float* kb    = qb    + R;
  float* vb    = kb    + R;
  float* hn    = base;                          // reuse after scan
  float* ob    = decay;                         // reuse after scan

  dm_kernel<<<TOK * 32 / 256, 256, 0, stream>>>(td, dm);
  input_gemm<<<TOK / 16, 256, 0, stream>>>(x, Win, bin, vt, dm,
                                           tef, tep, tdw, tdb, dcw, dcb,
                                           base, decay);
  scan_kernel<<<(B_ * H_) / 256, 256, 0, stream>>>(base, decay, lens, h);

  float* hcur = h;
  for (int l = 0; l < 2; ++l) {
    const float* Wq   = (const float*)d_in[LBASE + l*LSTRIDE + 0];
    const float* Wk   = (const float*)d_in[LBASE + l*LSTRIDE + 1];
    const float* Wv   = (const float*)d_in[LBASE + l*LSTRIDE + 2];
    const float* Wo   = (const float*)d_in[LBASE + l*LSTRIDE + 3];
    const float* bq   = (const float*)d_in[LBASE + l*LSTRIDE + 4];
    const float* bk   = (const float*)d_in[LBASE + l*LSTRIDE + 5];
    const float* bv   = (const float*)d_in[LBASE + l*LSTRIDE + 6];
    const float* bo   = (const float*)d_in[LBASE + l*LSTRIDE + 7];
    const float* wrel = (const float*)d_in[LBASE + l*LSTRIDE + 8];
    const float* lng  = (const float*)d_in[LBASE + l*LSTRIDE + 9];
    const float* lnb  = (const float*)d_in[LBASE + l*LSTRIDE + 10];

    ln_kernel<<<TOK * 32 / 256, 256, 0, stream>>>(hcur, lng, lnb, hn);
    gemm128<<<TOK / 16, 256, 0, stream>>>(hn, Wq, bq, nullptr, qb);
    gemm128<<<TOK / 16, 256, 0, stream>>>(hn, Wk, bk, nullptr, kb);
    gemm128<<<TOK / 16, 256, 0, stream>>>(hn, Wv, bv, nullptr, vb);

    attn_kernel<<<B_ * NH_ * (T_ / 16), 256,
                  ATTN_LDS_FLOATS * sizeof(float), stream>>>(
        qb, kb, vb, vt, lens, wrel, ob);

    float* hout = (l == 1) ? (float*)d_out : h;
    gemm128<<<TOK / 16, 256, 0, stream>>>(ob, Wo, bo, hcur, hout);
    hcur = hout;
  }
}